// SRenderY_9234179687452
// MI455X (gfx1250) — compile-verified
//
#include <hip/hip_runtime.h>

// ---------------------------------------------------------------------------
// SRenderY-style fused renderer for MI455X (gfx1250, wave32).
//
// Dead-code analysis of the reference: only attribute channels 0-1 (uv grid),
// 9-11 (fn = normals of `vertices`) and 12 (alpha) reach the output, so
// transformed_vertices / tfn / fv are never computed here.
//
// The SH shading einsum ('bsc,bshw->bchw') is a (3x9)x(9xN) GEMM; it is
// executed on the matrix pipe with V_WMMA_F32_16X16X4_F32: per wave, 2 tiles
// of 16 pixels (M) x 3 channels (N), K=9 padded to 12 (3 chained WMMAs).
// Operand layouts follow CDNA5 ISA 7.12.2 (32-bit A 16x4 / C,D 16x16).
// ---------------------------------------------------------------------------

#define Bc  4
#define Vc  5023
#define Fc  9976
#define Hc  512
#define UVc 256

typedef float v2f __attribute__((ext_vector_type(2)));
typedef float v8f __attribute__((ext_vector_type(8)));

// SH constants (match CONST in the reference)
#define SH_C0 0.28209479177387814f   // 1/sqrt(4*pi)
#define SH_C1 1.0233267079464885f    // 2*pi/3 * sqrt(3/(4*pi))
#define SH_C4 0.8580855308097834f    // pi/4 * 3 * sqrt(5/(12*pi))
#define SH_C7 0.4290427654048917f    // pi/4 * 1.5 * sqrt(5/(12*pi))
#define SH_C8 0.24770795610037571f   // pi/4 * 0.5 * sqrt(5/(4*pi))

__global__ __launch_bounds__(256) void k_zero(float* __restrict__ p, int n) {
    int i = blockIdx.x * 256 + threadIdx.x;
    if (i < n) p[i] = 0.0f;
}

// One thread per (batch, face): fn = cross(v1-v0, v2-v0), scatter-add to the
// 3 vertices (float atomics, same math as jnp .at[].add up to fp ordering).
__global__ __launch_bounds__(256) void k_facenorm(const float* __restrict__ verts,
                                                  const int*   __restrict__ faces,
                                                  float* __restrict__ vn) {
    int t = blockIdx.x * 256 + threadIdx.x;
    if (t >= Bc * Fc) return;
    int b = t / Fc;
    int f = t - b * Fc;
    int i0 = faces[f * 3 + 0];
    int i1 = faces[f * 3 + 1];
    int i2 = faces[f * 3 + 2];
    const float* v0 = verts + ((size_t)b * Vc + i0) * 3;
    const float* v1 = verts + ((size_t)b * Vc + i1) * 3;
    const float* v2 = verts + ((size_t)b * Vc + i2) * 3;
    float ax = v1[0] - v0[0], ay = v1[1] - v0[1], az = v1[2] - v0[2];
    float bx = v2[0] - v0[0], by = v2[1] - v0[1], bz = v2[2] - v0[2];
    float nx = ay * bz - az * by;
    float ny = az * bx - ax * bz;
    float nz = ax * by - ay * bx;
    float* p0 = vn + ((size_t)b * Vc + i0) * 3;
    float* p1 = vn + ((size_t)b * Vc + i1) * 3;
    float* p2 = vn + ((size_t)b * Vc + i2) * 3;
    atomicAdd(p0 + 0, nx); atomicAdd(p0 + 1, ny); atomicAdd(p0 + 2, nz);
    atomicAdd(p1 + 0, nx); atomicAdd(p1 + 1, ny); atomicAdd(p1 + 2, nz);
    atomicAdd(p2 + 0, nx); atomicAdd(p2 + 1, ny); atomicAdd(p2 + 2, nz);
}

__global__ __launch_bounds__(256) void k_normalize(float* __restrict__ vn) {
    int t = blockIdx.x * 256 + threadIdx.x;
    if (t >= Bc * Vc) return;
    float* p = vn + (size_t)t * 3;
    float x = p[0], y = p[1], z = p[2];
    float inv = 1.0f / fmaxf(sqrtf(x * x + y * y + z * z), 1e-6f);
    p[0] = x * inv; p[1] = y * inv; p[2] = z * inv;
}

// Fused: raster-interp (uv, normal, alpha) -> grid_sample -> SH shade (WMMA)
// -> multiply. One thread per pixel; grid is exactly B*H*H/256 full blocks so
// EXEC is all-ones at each WMMA (required by CDNA5 7.12).
__global__ __launch_bounds__(256) void k_main(const float* __restrict__ vnorm,    // (B,V,3) normalized
                                              const float* __restrict__ albedos,  // (B,3,UV,UV)
                                              const float* __restrict__ lights,   // (B,9,3)
                                              const float* __restrict__ uvcoords, // (1,F,3,3)
                                              const float* __restrict__ bary,     // (B,H,H,1,3)
                                              const int*   __restrict__ faces,    // (F,3)
                                              const int*   __restrict__ p2f,      // (B,H,H,1)
                                              float* __restrict__ out)            // (B,3,H,H)
{
    __shared__ float shS[256][12];    // per-pixel scaled SH basis, K padded 9->12
    __shared__ float shadeS[256][4];  // per-pixel shading, 3 channels (stride 4)

    const int tidL = threadIdx.x;
    const int gtid = blockIdx.x * 256 + tidL;
    const int pixPerImg = Hc * Hc;                 // 262144 (pow2)
    const int b   = gtid >> 18;                    // gtid / (H*H)
    const int pix = gtid & (pixPerImg - 1);

    // ---- rasterize_interp (channels 0-1, 9-11, alpha) ----
    int p = p2f[gtid];
    const float alpha = (p < 0) ? 0.0f : 1.0f;
    int idx = (p < 0) ? 0 : p;
    int b2 = idx / Fc;
    int f2 = idx - b2 * Fc;

    float w0 = bary[(size_t)gtid * 3 + 0];
    float w1 = bary[(size_t)gtid * 3 + 1];
    float w2 = bary[(size_t)gtid * 3 + 2];

    const float* uvf = uvcoords + (size_t)f2 * 9;   // (3 verts x 3)
    float gx = w0 * uvf[0] + w1 * uvf[3] + w2 * uvf[6];
    float gy = w0 * uvf[1] + w1 * uvf[4] + w2 * uvf[7];

    int i0 = faces[f2 * 3 + 0];
    int i1 = faces[f2 * 3 + 1];
    int i2 = faces[f2 * 3 + 2];
    const float* n0 = vnorm + ((size_t)b2 * Vc + i0) * 3;
    const float* n1 = vnorm + ((size_t)b2 * Vc + i1) * 3;
    const float* n2 = vnorm + ((size_t)b2 * Vc + i2) * 3;
    float Nx = w0 * n0[0] + w1 * n1[0] + w2 * n2[0];
    float Ny = w0 * n0[1] + w1 * n1[1] + w2 * n2[1];
    float Nz = w0 * n0[2] + w1 * n1[2] + w2 * n2[2];

    if (p < 0) { gx = 0.0f; gy = 0.0f; Nx = 0.0f; Ny = 0.0f; Nz = 0.0f; }

    // ---- grid_sample (bilinear, zero pad) on this pixel's own batch b ----
    float x = (gx + 1.0f) * (UVc * 0.5f) - 0.5f;
    float y = (gy + 1.0f) * (UVc * 0.5f) - 0.5f;
    float x0f = floorf(x), y0f = floorf(y);
    float wx = x - x0f,    wy = y - y0f;
    int x0 = (int)x0f, y0 = (int)y0f;
    int x1 = x0 + 1,   y1 = y0 + 1;
    float v00 = (x0 >= 0 && x0 < UVc && y0 >= 0 && y0 < UVc) ? 1.0f : 0.0f;
    float v01 = (x1 >= 0 && x1 < UVc && y0 >= 0 && y0 < UVc) ? 1.0f : 0.0f;
    float v10 = (x0 >= 0 && x0 < UVc && y1 >= 0 && y1 < UVc) ? 1.0f : 0.0f;
    float v11 = (x1 >= 0 && x1 < UVc && y1 >= 0 && y1 < UVc) ? 1.0f : 0.0f;
    int xc0 = min(max(x0, 0), UVc - 1), yc0 = min(max(y0, 0), UVc - 1);
    int xc1 = min(max(x1, 0), UVc - 1), yc1 = min(max(y1, 0), UVc - 1);
    float f00 = v00 * (1.0f - wx) * (1.0f - wy);
    float f01 = v01 * wx * (1.0f - wy);
    float f10 = v10 * (1.0f - wx) * wy;
    float f11 = v11 * wx * wy;
    float alb[3];
    #pragma unroll
    for (int c = 0; c < 3; ++c) {
        const float* img = albedos + ((size_t)(b * 3 + c)) * (UVc * UVc);
        float t00 = img[yc0 * UVc + xc0];
        float t01 = img[yc0 * UVc + xc1];
        float t10 = img[yc1 * UVc + xc0];
        float t11 = img[yc1 * UVc + xc1];
        alb[c] = t00 * f00 + t01 * f01 + t10 * f10 + t11 * f11;
    }

    // ---- SH basis (scaled by CONST), staged to LDS for the WMMA GEMM ----
    shS[tidL][0]  = SH_C0;
    shS[tidL][1]  = SH_C1 * Nx;
    shS[tidL][2]  = SH_C1 * Ny;
    shS[tidL][3]  = SH_C1 * Nz;
    shS[tidL][4]  = SH_C4 * Nx * Ny;
    shS[tidL][5]  = SH_C4 * Nx * Nz;
    shS[tidL][6]  = SH_C4 * Ny * Nz;
    shS[tidL][7]  = SH_C7 * (Nx * Nx - Ny * Ny);
    shS[tidL][8]  = SH_C8 * (3.0f * Nz * Nz - 1.0f);
    shS[tidL][9]  = 0.0f;
    shS[tidL][10] = 0.0f;
    shS[tidL][11] = 0.0f;
    __syncthreads();

    // ---- shading^T[pixel][chan] = sum_k shS[pixel][k] * lights[b][k][chan]
    //      via V_WMMA_F32_16X16X4_F32: M=16 pixels, N=16 (3 used), K=4 x3 ----
    const int lane     = tidL & 31;
    const int waveBase = tidL & ~31;
    const int m        = lane & 15;             // A: row M; B: col N
    const int khalf    = (lane < 16) ? 0 : 2;   // K pair held by this lane

    v2f Bop[3];                                 // B operand per K-chunk (tile-invariant)
    #pragma unroll
    for (int ch = 0; ch < 3; ++ch) {
        int k0 = ch * 4 + khalf;
        int k1 = k0 + 1;
        float b0 = (m < 3 && k0 < 9) ? lights[((size_t)b * 9 + k0) * 3 + m] : 0.0f;
        float b1 = (m < 3 && k1 < 9) ? lights[((size_t)b * 9 + k1) * 3 + m] : 0.0f;
        Bop[ch].x = b0;
        Bop[ch].y = b1;
    }

    #pragma unroll
    for (int t = 0; t < 2; ++t) {               // 2 tiles of 16 pixels per wave
        const int tileBase = waveBase + t * 16;
        v8f Cacc = {};
        #pragma unroll
        for (int ch = 0; ch < 3; ++ch) {
            v2f A;
            A.x = shS[tileBase + m][ch * 4 + khalf];
            A.y = shS[tileBase + m][ch * 4 + khalf + 1];
            Cacc = __builtin_amdgcn_wmma_f32_16x16x4_f32(
                false, A, false, Bop[ch], (short)0, Cacc, false, false);
        }
        // D layout (32-bit 16x16): lane l, VGPR r -> M = r + (l<16?0:8), N = l&15
        if (m < 3) {
            #pragma unroll
            for (int r = 0; r < 8; ++r) {
                int mm = r + ((lane < 16) ? 0 : 8);
                shadeS[tileBase + mm][m] = Cacc[r];
            }
        }
    }
    __syncthreads();

    // ---- images = albedo * shading * alpha ----
    size_t obase = (size_t)b * 3 * pixPerImg + pix;
    out[obase]                   = alb[0] * shadeS[tidL][0] * alpha;
    out[obase + pixPerImg]       = alb[1] * shadeS[tidL][1] * alpha;
    out[obase + 2 * pixPerImg]   = alb[2] * shadeS[tidL][2] * alpha;
}

extern "C" void kernel_launch(void* const* d_in, const int* in_sizes, int n_in,
                              void* d_out, int out_size, void* d_ws, size_t ws_size,
                              hipStream_t stream) {
    (void)in_sizes; (void)n_in; (void)out_size; (void)ws_size;
    const float* vertices = (const float*)d_in[0];
    // d_in[1] transformed_vertices: dead (channels 3-8 never reach the output)
    const float* albedos  = (const float*)d_in[2];
    const float* lights   = (const float*)d_in[3];
    const float* uvcoords = (const float*)d_in[4];
    const float* bary     = (const float*)d_in[5];
    const int*   faces    = (const int*)d_in[6];
    const int*   p2f      = (const int*)d_in[7];
    float*       out      = (float*)d_out;

    float* vn = (float*)d_ws;                 // (B,V,3) accumulator, 241 KB
    const int nvn = Bc * Vc * 3;

    k_zero<<<(nvn + 255) / 256, 256, 0, stream>>>(vn, nvn);
    k_facenorm<<<(Bc * Fc + 255) / 256, 256, 0, stream>>>(vertices, faces, vn);
    k_normalize<<<(Bc * Vc + 255) / 256, 256, 0, stream>>>(vn);

    const int npix = Bc * Hc * Hc;            // 1048576, exact multiple of 256
    k_main<<<npix / 256, 256, 0, stream>>>(vn, albedos, lights, uvcoords,
                                           bary, faces, p2f, out);
}